// GNN_65807488909362
// MI455X (gfx1250) — compile-verified
//
#include <hip/hip_runtime.h>
#include <hip/hip_bf16.h>

typedef __attribute__((ext_vector_type(2))) float v2f;
typedef __attribute__((ext_vector_type(8))) float v8f;

#define TB 256

// ---------------- threefry2x32 (JAX-compatible, key = (0, 42)) ----------------
#define TF_ROUND(r) { x0 += x1; x1 = (x1 << (r)) | (x1 >> (32 - (r))); x1 ^= x0; }

__device__ __forceinline__ unsigned threefry_bits(unsigned i, unsigned halfn) {
    unsigned x0, x1; bool lo;
    if (i < halfn) { x0 = i;         x1 = i + halfn; lo = true; }
    else           { x0 = i - halfn; x1 = i;         lo = false; }
    const unsigned k0 = 0u, k1 = 42u;
    const unsigned k2 = k0 ^ k1 ^ 0x1BD11BDAu;
    x0 += k0; x1 += k1;
    TF_ROUND(13) TF_ROUND(15) TF_ROUND(26) TF_ROUND(6)
    x0 += k1; x1 += k2 + 1u;
    TF_ROUND(17) TF_ROUND(29) TF_ROUND(16) TF_ROUND(24)
    x0 += k2; x1 += k0 + 2u;
    TF_ROUND(13) TF_ROUND(15) TF_ROUND(26) TF_ROUND(6)
    x0 += k0; x1 += k1 + 3u;
    TF_ROUND(17) TF_ROUND(29) TF_ROUND(16) TF_ROUND(24)
    x0 += k1; x1 += k2 + 4u;
    TF_ROUND(13) TF_ROUND(15) TF_ROUND(26) TF_ROUND(6)
    x0 += k2; x1 += k0 + 5u;
    return lo ? x0 : x1;
}

// monotone float <-> uint key (for atomicMax-based float max)
__device__ __forceinline__ unsigned f2key(float f) {
    unsigned u = __float_as_uint(f);
    return (u & 0x80000000u) ? ~u : (u | 0x80000000u);
}
__device__ __forceinline__ float key2f(unsigned k) {
    unsigned u = (k & 0x80000000u) ? (k & 0x7FFFFFFFu) : ~k;
    return __uint_as_float(u);
}

// ---------------- edge normalization ----------------
__global__ void deg_kernel(const int* __restrict__ src, const int* __restrict__ dst,
                           const float* __restrict__ w, float* __restrict__ deg, int E) {
    int e = blockIdx.x * blockDim.x + threadIdx.x;
    if (e >= E) return;
    int s = src[e];
    float wv = (s == dst[e]) ? 0.0f : w[e];
    if (wv != 0.0f) atomicAdd(&deg[s], wv);
}

__global__ void dis_kernel(float* __restrict__ deg, int N) {
    int i = blockIdx.x * blockDim.x + threadIdx.x;
    if (i >= N) return;
    float d = deg[i];
    deg[i] = (d > 0.0f) ? rsqrtf(d) : 0.0f;
}

__global__ void norm_kernel(const int* __restrict__ src, const int* __restrict__ dst,
                            const float* __restrict__ w, const float* __restrict__ dis,
                            float* __restrict__ norm, int E) {
    int e = blockIdx.x * blockDim.x + threadIdx.x;
    if (e >= E) return;
    int s = src[e], d = dst[e];
    float wv = (s == d) ? 0.0f : w[e];
    norm[e] = -dis[s] * wv * dis[d];
}

// ---------------- weighted edge scatter: out[dst] += norm * in[src], 16 channels ----
__global__ void scatter_kernel(const float* __restrict__ in, float* __restrict__ out,
                               const int* __restrict__ src, const int* __restrict__ dst,
                               const float* __restrict__ norm, int E) {
    int e = blockIdx.x * blockDim.x + threadIdx.x;
    if (e >= E) return;
    float w = norm[e];
    if (w == 0.0f) return;
    int s = src[e], d = dst[e];
    const float4* ip = (const float4*)(in + (size_t)s * 16);
    float* op = out + (size_t)d * 16;
#pragma unroll
    for (int q = 0; q < 4; ++q) {
        float4 a = ip[q];
        atomicAdd(op + q * 4 + 0, w * a.x);
        atomicAdd(op + q * 4 + 1, w * a.y);
        atomicAdd(op + q * 4 + 2, w * a.z);
        atomicAdd(op + q * 4 + 3, w * a.w);
    }
}

// ---------------- WMMA combine: out = act( Tx0@W0 + Tx1@W1 + (2*t2 - Tx0)@W2 + b ) ----
// One wave per 16-node tile. 12 chained v_wmma_f32_16x16x4_f32 per tile.
__global__ __launch_bounds__(TB)
void combine16_wmma(const float* __restrict__ t0, const float* __restrict__ t1,
                    const float* __restrict__ t2, const float* __restrict__ W,   // [3][16][16]
                    const float* __restrict__ bias,                               // [16]
                    float* __restrict__ out, int n_nodes,
                    int do_relu, int do_dropout, unsigned halfn) {
    int wave = blockIdx.x * (blockDim.x >> 5) + (threadIdx.x >> 5);
    int lane = threadIdx.x & 31;
    int tile = wave;
    if (tile * 16 >= n_nodes) return;           // wave-uniform: EXEC stays all-ones
    int half = lane >> 4;                        // K/M half selector
    int l15  = lane & 15;
    int row  = tile * 16 + l15;                  // A-matrix row owned by this lane

    const float* a0 = t0 + (size_t)row * 16;
    const float* a1 = t1 + (size_t)row * 16;
    const float* a2 = t2 + (size_t)row * 16;

    v8f acc = {};
#pragma unroll
    for (int k = 0; k < 3; ++k) {
        const float* ak = (k == 0) ? a0 : ((k == 1) ? a1 : a2);
        const float* wk = W + k * 256;
#pragma unroll
        for (int kk = 0; kk < 4; ++kk) {
            int col = kk * 4 + half * 2;         // K columns this lane supplies
            v2f a, b;
            if (k < 2) {
                a.x = ak[col];
                a.y = ak[col + 1];
            } else {                              // Tx2 = 2*lhat(Tx1) - Tx0, fused here
                a.x = 2.0f * ak[col]     - a0[col];
                a.y = 2.0f * ak[col + 1] - a0[col + 1];
            }
            b.x = wk[(col    ) * 16 + l15];       // B rows striped across lanes
            b.y = wk[(col + 1) * 16 + l15];
            acc = __builtin_amdgcn_wmma_f32_16x16x4_f32(
                false, a, false, b, (short)0, acc, false, false);
        }
    }

    float bv = bias[l15];
#pragma unroll
    for (int v = 0; v < 8; ++v) {
        int m    = v + half * 8;                  // C/D layout: VGPR v -> rows v / v+8
        int node = tile * 16 + m;
        float val = acc[v] + bv;
        if (do_relu) val = fmaxf(val, 0.0f);
        if (do_dropout) {
            unsigned bits = threefry_bits((unsigned)(node * 16 + l15), halfn);
            val = (bits & 0x80000000u) ? 0.0f : val * 2.0f;   // uniform<0.5 <=> MSB==0
        }
        out[(size_t)node * 16 + l15] = val;
    }
}

// ---------------- layer 3: scalar channel out ----------------
__global__ void layer3_kernel(const float* __restrict__ h2, const float* __restrict__ t1,
                              const float* __restrict__ t2, const float* __restrict__ W3, // [3][16]
                              const float* __restrict__ b3, float* __restrict__ c, int N) {
    int n = blockIdx.x * blockDim.x + threadIdx.x;
    if (n >= N) return;
    const float* r0 = h2 + (size_t)n * 16;
    const float* r1 = t1 + (size_t)n * 16;
    const float* r2 = t2 + (size_t)n * 16;
    float acc = b3[0];
#pragma unroll
    for (int d = 0; d < 16; ++d) {
        acc += r0[d] * W3[d] + r1[d] * W3[16 + d] + (2.0f * r2[d] - r0[d]) * W3[32 + d];
    }
    c[n] = acc;
}

// ---------------- column sum of h2 (for global mean pool) ----------------
__global__ void colsum_kernel(const float* __restrict__ h, float* __restrict__ vsum, int N) {
    __shared__ float sh[TB * 16];
    float loc[16];
#pragma unroll
    for (int cc = 0; cc < 16; ++cc) loc[cc] = 0.0f;
    for (int i = blockIdx.x * blockDim.x + threadIdx.x; i < N; i += gridDim.x * blockDim.x) {
        const float4* p = (const float4*)(h + (size_t)i * 16);
#pragma unroll
        for (int q = 0; q < 4; ++q) {
            float4 a = p[q];
            loc[q * 4 + 0] += a.x; loc[q * 4 + 1] += a.y;
            loc[q * 4 + 2] += a.z; loc[q * 4 + 3] += a.w;
        }
    }
#pragma unroll
    for (int cc = 0; cc < 16; ++cc) sh[threadIdx.x * 16 + cc] = loc[cc];
    __syncthreads();
    for (int s = TB / 2; s > 0; s >>= 1) {
        if ((int)threadIdx.x < s) {
#pragma unroll
            for (int cc = 0; cc < 16; ++cc)
                sh[threadIdx.x * 16 + cc] += sh[(threadIdx.x + s) * 16 + cc];
        }
        __syncthreads();
    }
    if (threadIdx.x < 16) atomicAdd(&vsum[threadIdx.x], sh[threadIdx.x]);
}

// ---------------- softmax over nodes ----------------
__global__ void max_kernel(const float* __restrict__ c, unsigned* __restrict__ maxkey, int N) {
    __shared__ unsigned sh[TB];
    unsigned k = 0u;
    for (int i = blockIdx.x * blockDim.x + threadIdx.x; i < N; i += gridDim.x * blockDim.x) {
        unsigned t = f2key(c[i]);
        if (t > k) k = t;
    }
    sh[threadIdx.x] = k;
    __syncthreads();
    for (int s = TB / 2; s > 0; s >>= 1) {
        if ((int)threadIdx.x < s) {
            unsigned o = sh[threadIdx.x + s];
            if (o > sh[threadIdx.x]) sh[threadIdx.x] = o;
        }
        __syncthreads();
    }
    if (threadIdx.x == 0) atomicMax(maxkey, sh[0]);
}

__global__ void expsum_kernel(float* __restrict__ c, const unsigned* __restrict__ maxkey,
                              float* __restrict__ sum, int N) {
    __shared__ float sh[TB];
    float mx = key2f(*maxkey);
    float acc = 0.0f;
    for (int i = blockIdx.x * blockDim.x + threadIdx.x; i < N; i += gridDim.x * blockDim.x) {
        float e = expf(c[i] - mx);
        c[i] = e;
        acc += e;
    }
    sh[threadIdx.x] = acc;
    __syncthreads();
    for (int s = TB / 2; s > 0; s >>= 1) {
        if ((int)threadIdx.x < s) sh[threadIdx.x] += sh[threadIdx.x + s];
        __syncthreads();
    }
    if (threadIdx.x == 0) atomicAdd(sum, sh[0]);
}

__global__ void choice_kernel(const float* __restrict__ c, const float* __restrict__ sum,
                              float* __restrict__ out, int N) {
    int n = blockIdx.x * blockDim.x + threadIdx.x;
    if (n >= N) return;
    out[n] = c[n] / (*sum);
}

__global__ void value_kernel(const float* __restrict__ vsum, const float* __restrict__ Wv,
                             const float* __restrict__ bv, float* __restrict__ out, int N) {
    float acc = 0.0f;
    float inv = 1.0f / (float)N;
#pragma unroll
    for (int cc = 0; cc < 16; ++cc) acc += (vsum[cc] * inv) * Wv[cc];
    out[0] = acc + bv[0];
}

// =============================================================================
extern "C" void kernel_launch(void* const* d_in, const int* in_sizes, int n_in,
                              void* d_out, int out_size, void* d_ws, size_t ws_size,
                              hipStream_t stream) {
    const float* x      = (const float*)d_in[0];
    const int*   eidx   = (const int*)  d_in[1];
    const float* weight = (const float*)d_in[2];
    const float* W1 = (const float*)d_in[3];
    const float* b1 = (const float*)d_in[4];
    const float* W2 = (const float*)d_in[5];
    const float* b2 = (const float*)d_in[6];
    const float* W3 = (const float*)d_in[7];
    const float* b3 = (const float*)d_in[8];
    const float* Wv = (const float*)d_in[9];
    const float* bv = (const float*)d_in[10];

    const int N = in_sizes[0] / 16;
    const int E = in_sizes[2];
    const int* src = eidx;
    const int* dst = eidx + E;
    const unsigned halfn = (unsigned)((N * 16) / 2);   // threefry counter split point

    // workspace carve-up (256B aligned)
    char* ws = (char*)d_ws;
    size_t off = 0;
    auto take = [&](size_t bytes) -> char* {
        char* p = ws + off;
        off = (off + bytes + 255) & ~(size_t)255;
        return p;
    };
    float*    norm = (float*)take((size_t)E * 4);
    float*    dis  = (float*)take((size_t)N * 4);        // degree -> deg^-1/2 in place
    float*    t1   = (float*)take((size_t)N * 64);
    float*    t2   = (float*)take((size_t)N * 64);
    float*    h1   = (float*)take((size_t)N * 64);
    float*    h2   = (float*)take((size_t)N * 64);
    float*    cbuf = (float*)take((size_t)N * 4);
    float*    vsum = (float*)take(64);
    unsigned* scal = (unsigned*)take(32);                // [0]=maxkey, [1]=sumexp
    float*    sump = (float*)(scal + 1);

    float* out_choice = (float*)d_out;
    float* out_value  = (float*)d_out + N;

    const int gE = (E + TB - 1) / TB;
    const int gN = (N + TB - 1) / TB;
    const int waves = (N + 15) / 16;
    const int gW = (waves + (TB / 32) - 1) / (TB / 32);
    const int gR = 512;                                  // grid-stride reductions

    // ---- edge normalization ----
    hipMemsetAsync(dis, 0, (size_t)N * 4, stream);
    deg_kernel <<<gE, TB, 0, stream>>>(src, dst, weight, dis, E);
    dis_kernel <<<gN, TB, 0, stream>>>(dis, N);
    norm_kernel<<<gE, TB, 0, stream>>>(src, dst, weight, dis, norm, E);

    // ---- layer 1: x -> h1 (relu + dropout) ----
    hipMemsetAsync(t1, 0, (size_t)N * 64, stream);
    scatter_kernel<<<gE, TB, 0, stream>>>(x, t1, src, dst, norm, E);
    hipMemsetAsync(t2, 0, (size_t)N * 64, stream);
    scatter_kernel<<<gE, TB, 0, stream>>>(t1, t2, src, dst, norm, E);
    combine16_wmma<<<gW, TB, 0, stream>>>(x, t1, t2, W1, b1, h1, N, 1, 1, halfn);

    // ---- layer 2: h1 -> h2 (relu) ----
    hipMemsetAsync(t1, 0, (size_t)N * 64, stream);
    scatter_kernel<<<gE, TB, 0, stream>>>(h1, t1, src, dst, norm, E);
    hipMemsetAsync(t2, 0, (size_t)N * 64, stream);
    scatter_kernel<<<gE, TB, 0, stream>>>(t1, t2, src, dst, norm, E);
    combine16_wmma<<<gW, TB, 0, stream>>>(h1, t1, t2, W2, b2, h2, N, 1, 0, halfn);

    // ---- layer 3: h2 -> c [N,1] ----
    hipMemsetAsync(t1, 0, (size_t)N * 64, stream);
    scatter_kernel<<<gE, TB, 0, stream>>>(h2, t1, src, dst, norm, E);
    hipMemsetAsync(t2, 0, (size_t)N * 64, stream);
    scatter_kernel<<<gE, TB, 0, stream>>>(t1, t2, src, dst, norm, E);
    layer3_kernel<<<gN, TB, 0, stream>>>(h2, t1, t2, W3, b3, cbuf, N);

    // ---- outputs: softmax(c) over nodes, value = mean(h2) @ Wv.T + bv ----
    hipMemsetAsync(scal, 0, 32, stream);                 // maxkey=0 (== -inf key), sum=0
    hipMemsetAsync(vsum, 0, 64, stream);
    colsum_kernel<<<gR, TB, 0, stream>>>(h2, vsum, N);
    max_kernel   <<<gR, TB, 0, stream>>>(cbuf, scal, N);
    expsum_kernel<<<gR, TB, 0, stream>>>(cbuf, scal, sump, N);
    choice_kernel<<<gN, TB, 0, stream>>>(cbuf, sump, out_choice, N);
    value_kernel <<<1, 1, 0, stream>>>(vsum, Wv, bv, out_value, N);
}